// NMSLayer_39505109189245
// MI455X (gfx1250) — compile-verified
//
#include <hip/hip_runtime.h>

#define NIMG 16
#define IMGE (1024 * 1024)
#define IMG_W 1024
#define IMG_H 1024
#define MEDK ((IMGE - 1) / 2)   // lower median rank = 524287
#define CAP  8192               // per-image collected-key capacity (~100x margin)

// ---------- CDNA5 async global->LDS path (probe with __has_builtin) ----------
#ifndef USE_ASYNC_LDS
#if defined(__has_builtin)
#if __has_builtin(__builtin_amdgcn_global_load_async_to_lds_b128) && \
    __has_builtin(__builtin_amdgcn_s_wait_asynccnt)
#define USE_ASYNC_LDS 1
#endif
#endif
#endif
#ifndef USE_ASYNC_LDS
#define USE_ASYNC_LDS 0
#endif

#if USE_ASYNC_LDS
typedef int v4i __attribute__((vector_size(16)));
typedef __attribute__((address_space(1))) v4i g_v4i;  // global (cuda_device)
typedef __attribute__((address_space(3))) v4i l_v4i;  // LDS (cuda_shared)
#endif

// Monotonic key: uint compare order == float compare order (finite data).
__device__ __forceinline__ unsigned mono(float f) {
  unsigned u = __float_as_uint(f);
  return (u & 0x80000000u) ? ~u : (u | 0x80000000u);
}

// threshold: x >= med -> x, else 0; -inf (tile padding) passes through as -inf
__device__ __forceinline__ float thresh(float v, unsigned mm) {
  unsigned u = __float_as_uint(v);
  unsigned m = (u & 0x80000000u) ? ~u : (u | 0x80000000u);
  float z = (u == 0xFF800000u) ? v : 0.0f;  // keep -inf padding as identity
  return (m >= mm) ? v : z;
}

// -------- Fused threshold + separable 7x7 max-pool + NMS compare ------------
// Placed first so the disasm snippet shows the async-to-LDS path.
#define TOX 128      // output tile width
#define TOY 32       // output tile height
#define LH  38       // loaded height (TOY + 6)
#define PW  136      // loaded width in floats = 34 float4 (covers halo, aligned)
#define LW4 34       // float4 loads per row

__global__ __launch_bounds__(256) void k_nms(const float* __restrict__ x,
                                             const unsigned* __restrict__ medm,
                                             float* __restrict__ out) {
  __shared__ __align__(16) float tile[LH][PW];
  __shared__ float hm[LH][TOX];
  const int tid = threadIdx.x;
  const int img = blockIdx.z;
  const int gx0 = blockIdx.x * TOX - 4;   // float4-aligned tile origin
  const int gy0 = blockIdx.y * TOY - 3;
  const float* ip = x + (size_t)img * IMGE;
  float* op = out + (size_t)img * IMGE;

  // Prefill with -inf (reference reduce_window -inf init / padding).
  float* t1 = &tile[0][0];
  for (int i = tid; i < LH * PW; i += 256) t1[i] = -__builtin_inff();
  __syncthreads();

  // Stage raw halo tile into LDS with async B128 copies (ASYNCcnt tracked).
  // Every float4 is fully in-image or fully out (1024 % 4 == 0, gx0 % 4 == 0).
  for (int i = tid; i < LH * LW4; i += 256) {
    int ty = i / LW4, f4 = i - ty * LW4;
    int gy = gy0 + ty;
    int gxb = gx0 + f4 * 4;
    if ((unsigned)gy < (unsigned)IMG_H && (unsigned)gxb < (unsigned)IMG_W) {
#if USE_ASYNC_LDS
      __builtin_amdgcn_global_load_async_to_lds_b128(
          (g_v4i*)(ip + (size_t)gy * IMG_W + gxb),
          (l_v4i*)&tile[ty][f4 * 4], 0, 0);
#else
      *(float4*)&tile[ty][f4 * 4] = *(const float4*)(ip + (size_t)gy * IMG_W + gxb);
#endif
    }
  }
#if USE_ASYNC_LDS
  __builtin_amdgcn_s_wait_asynccnt(0);
#endif
  __syncthreads();

  // Threshold in place (keeps -inf padding).
  const unsigned mm = medm[img];
  for (int i = tid; i < LH * PW; i += 256) t1[i] = thresh(t1[i], mm);
  __syncthreads();

  // Horizontal 7-tap max: window for output col xx is tile cols [xx+1, xx+7].
  for (int i = tid; i < LH * TOX; i += 256) {
    int y = i >> 7, xx = i & 127;
    float m = tile[y][xx + 1];
#pragma unroll
    for (int d = 2; d < 8; ++d) m = fmaxf(m, tile[y][xx + d]);
    hm[y][xx] = m;
  }
  __syncthreads();

  // Vertical 7-tap max + NMS keep + float4 store.
  for (int i = tid; i < TOY * (TOX / 4); i += 256) {
    int y = i >> 5, x4 = (i & 31) * 4;
    float4 r;
    float* rp = &r.x;
#pragma unroll
    for (int j = 0; j < 4; ++j) {
      int xx = x4 + j;
      float m = hm[y][xx];
#pragma unroll
      for (int d = 1; d < 7; ++d) m = fmaxf(m, hm[y + d][xx]);
      float c = tile[y + 3][xx + 4];
      rp[j] = (c == m) ? c : 0.0f;
    }
    *(float4*)(op + (size_t)(gy0 + 3 + y) * IMG_W + (gx0 + 4 + x4)) = r;
  }
}

// ---------------- Pass 1: 4096-bin histogram of mono>>20 (LDS privatized) ---
__global__ __launch_bounds__(256) void k_hist1(const float* __restrict__ x,
                                               unsigned* __restrict__ hist) {
  __shared__ unsigned h[4096];
  for (int i = threadIdx.x; i < 4096; i += 256) h[i] = 0u;
  __syncthreads();
  const int img = blockIdx.y;
  const size_t base = (size_t)img * IMGE + (size_t)blockIdx.x * 16384;
  const float4* p = (const float4*)(x + base);
  for (int j = 0; j < 16; ++j) {
    float4 v = p[j * 256 + threadIdx.x];
    atomicAdd(&h[mono(v.x) >> 20], 1u);
    atomicAdd(&h[mono(v.y) >> 20], 1u);
    atomicAdd(&h[mono(v.z) >> 20], 1u);
    atomicAdd(&h[mono(v.w) >> 20], 1u);
  }
  __syncthreads();
  unsigned* hg = hist + (size_t)img * 4096;
  for (int i = threadIdx.x; i < 4096; i += 256) {
    unsigned c = h[i];
    if (c) atomicAdd(&hg[i], c);
  }
}

// ------------- Serial bin scan: find bin holding rank k, remaining rank -----
__global__ void k_select(const unsigned* __restrict__ hist, int nbins,
                         unsigned* __restrict__ bin_out,
                         unsigned* __restrict__ rank_out) {
  if (threadIdx.x != 0) return;
  const int img = blockIdx.x;
  const unsigned k = (unsigned)MEDK;
  const unsigned* h = hist + (size_t)img * nbins;
  unsigned cum = 0;
  for (int i = 0; i < nbins; ++i) {
    unsigned c = h[i];
    if (cum + c > k) {
      bin_out[img] = (unsigned)i;
      rank_out[img] = k - cum;
      return;
    }
    cum += c;
  }
  bin_out[img] = (unsigned)(nbins - 1);
  rank_out[img] = 0u;
}

// ---- Pass 2: collect 20-bit key suffixes of elements in the winning bin ----
__global__ __launch_bounds__(256) void k_collect(const float* __restrict__ x,
                                                 const unsigned* __restrict__ b1,
                                                 unsigned* __restrict__ buf,
                                                 unsigned* __restrict__ cnt) {
  const int img = blockIdx.y;
  const unsigned top = b1[img];
  const size_t base = (size_t)img * IMGE + (size_t)blockIdx.x * 16384;
  const float4* p = (const float4*)(x + base);
  unsigned* bb = buf + (size_t)img * CAP;
  for (int j = 0; j < 16; ++j) {
    float4 v = p[j * 256 + threadIdx.x];
    unsigned u;
    u = mono(v.x);
    if ((u >> 20) == top) { unsigned id = atomicAdd(&cnt[img], 1u); if (id < CAP) bb[id] = u & 0xFFFFFu; }
    u = mono(v.y);
    if ((u >> 20) == top) { unsigned id = atomicAdd(&cnt[img], 1u); if (id < CAP) bb[id] = u & 0xFFFFFu; }
    u = mono(v.z);
    if ((u >> 20) == top) { unsigned id = atomicAdd(&cnt[img], 1u); if (id < CAP) bb[id] = u & 0xFFFFFu; }
    u = mono(v.w);
    if ((u >> 20) == top) { unsigned id = atomicAdd(&cnt[img], 1u); if (id < CAP) bb[id] = u & 0xFFFFFu; }
  }
}

// ---- Final select: two-level 10+10-bit histogram over collected keys -------
__global__ __launch_bounds__(256) void k_sel2(const unsigned* __restrict__ buf,
                                              const unsigned* __restrict__ cnt,
                                              const unsigned* __restrict__ b1,
                                              const unsigned* __restrict__ k1,
                                              unsigned* __restrict__ medm) {
  __shared__ unsigned h[1024];
  __shared__ unsigned sel;
  const int img = blockIdx.x;
  unsigned n = cnt[img];
  if (n > (unsigned)CAP) n = (unsigned)CAP;
  const unsigned* b = buf + (size_t)img * CAP;
  const unsigned target = k1[img];

  for (int i = threadIdx.x; i < 1024; i += 256) h[i] = 0u;
  __syncthreads();
  for (unsigned i = threadIdx.x; i < n; i += 256)
    atomicAdd(&h[(b[i] >> 10) & 0x3FFu], 1u);
  __syncthreads();
  if (threadIdx.x == 0) {
    unsigned cum = 0, bin = 1023u, rem = 0u;
    for (int i = 0; i < 1024; ++i) {
      unsigned c = h[i];
      if (cum + c > target) { bin = (unsigned)i; rem = target - cum; break; }
      cum += c;
    }
    sel = (bin << 16) | rem;   // rem < CAP = 8192 fits in 16 bits
  }
  __syncthreads();
  const unsigned bin_hi = sel >> 16;
  const unsigned rem = sel & 0xFFFFu;

  for (int i = threadIdx.x; i < 1024; i += 256) h[i] = 0u;
  __syncthreads();
  for (unsigned i = threadIdx.x; i < n; i += 256) {
    unsigned u = b[i];
    if (((u >> 10) & 0x3FFu) == bin_hi) atomicAdd(&h[u & 0x3FFu], 1u);
  }
  __syncthreads();
  if (threadIdx.x == 0) {
    unsigned cum = 0, lo = 1023u;
    for (int i = 0; i < 1024; ++i) {
      unsigned c = h[i];
      if (cum + c > rem) { lo = (unsigned)i; break; }
      cum += c;
    }
    medm[img] = (b1[img] << 20) | (bin_hi << 10) | lo;
  }
}

// ---------------------------------------------------------------------------
extern "C" void kernel_launch(void* const* d_in, const int* in_sizes, int n_in,
                              void* d_out, int out_size, void* d_ws, size_t ws_size,
                              hipStream_t stream) {
  (void)in_sizes; (void)n_in; (void)out_size; (void)ws_size;
  const float* x = (const float*)d_in[0];
  float* out = (float*)d_out;

  unsigned* hist1 = (unsigned*)d_ws;            // 16 * 4096
  unsigned* cnt = hist1 + NIMG * 4096;          // 16   (zeroed with hist1)
  unsigned* b1 = cnt + NIMG;                    // 16
  unsigned* k1 = b1 + NIMG;                     // 16
  unsigned* mm = k1 + NIMG;                     // 16
  unsigned* buf = mm + NIMG;                    // 16 * CAP

  const size_t zbytes = (size_t)(NIMG * 4096 + NIMG) * sizeof(unsigned);
  (void)hipMemsetAsync(d_ws, 0, zbytes, stream);  // graph-capturable

  k_hist1<<<dim3(64, NIMG), 256, 0, stream>>>(x, hist1);
  k_select<<<NIMG, 32, 0, stream>>>(hist1, 4096, b1, k1);
  k_collect<<<dim3(64, NIMG), 256, 0, stream>>>(x, b1, buf, cnt);
  k_sel2<<<NIMG, 256, 0, stream>>>(buf, cnt, b1, k1, mm);
  k_nms<<<dim3(IMG_W / TOX, IMG_H / TOY, NIMG), 256, 0, stream>>>(x, mm, out);
}